// PointLSTMEncoder_30932354466225
// MI455X (gfx1250) — compile-verified
//
#include <hip/hip_runtime.h>
#include <hip/hip_bf16.h>
#include <stddef.h>

// Problem constants (from reference)
#define B_    2
#define T_    16
#define CIN   132
#define NPTS  512
#define KNN_  16
#define HID   256
#define CTOT  392      // CIN + 4 + HID
#define OUTC  260      // 4 + HID

// GEMM tiling
#define HTILE   32      // hidden units per workgroup  -> 128 W-rows (4 gates x 32)
#define COLT    64      // columns per workgroup (4 points x 16 neighbors)
#define KC      28      // k-chunk (7 wmma k-steps); 392 = 14 * 28
#define NKC     14
#define ASTRIDE 30      // even stride -> 8B-aligned ds_load_b64 fragments
#define BSOFF   (128 * ASTRIDE)          // 3840 floats
#define GSTRIDE 66
#define SMEMF   (128 * GSTRIDE)          // 8448 floats = 33792 B shared

typedef __attribute__((ext_vector_type(2))) float v2f;
typedef __attribute__((ext_vector_type(8))) float v8f;

__device__ __forceinline__ float sigf(float x)  { return 1.0f / (1.0f + __expf(-x)); }
__device__ __forceinline__ float tanhf_(float x){ return 1.0f - 2.0f / (1.0f + __expf(2.0f * x)); }

// ---------------------------------------------------------------------------
// KNN: for frame t, neighbors of p3[:,t] among p3[:,t-1] (t=0: among itself).
// One block per (b, t, half); thread = one query row; candidates staged in LDS.
// ---------------------------------------------------------------------------
__global__ __launch_bounds__(256) void knn_kernel(const float* __restrict__ x,
                                                  int* __restrict__ idxbuf) {
    __shared__ float cx[NPTS], cy[NPTS], cz[NPTS];
    const int bt = blockIdx.x >> 1, half = blockIdx.x & 1;
    const int b = bt / T_, t = bt % T_;
    const int tprev = (t == 0) ? 0 : (t - 1);
    const float* xc = x + (size_t)((b * T_ + tprev) * CIN) * NPTS;
    for (int i = threadIdx.x; i < NPTS; i += blockDim.x) {
        cx[i] = xc[i];
        cy[i] = xc[NPTS + i];
        cz[i] = xc[2 * NPTS + i];
    }
    __syncthreads();

    const int row = half * 256 + threadIdx.x;
    const float* xq = x + (size_t)((b * T_ + t) * CIN) * NPTS;
    const float qx = xq[row], qy = xq[NPTS + row], qz = xq[2 * NPTS + row];

    float bd[KNN_];
    int   bi[KNN_];
#pragma unroll
    for (int s = 0; s < KNN_; s++) { bd[s] = 3.4e38f; bi[s] = 0; }

    for (int j = 0; j < NPTS; j++) {
        const float dx = qx - cx[j], dy = qy - cy[j], dz = qz - cz[j];
        const float d2 = dx * dx + dy * dy + dz * dz;
        if (d2 < bd[KNN_ - 1]) {            // strict <: ties keep lower index (jax top_k)
            bd[KNN_ - 1] = d2; bi[KNN_ - 1] = j;
#pragma unroll
            for (int s = KNN_ - 1; s >= 1; --s) {
                if (bd[s] < bd[s - 1]) {
                    float td = bd[s]; bd[s] = bd[s - 1]; bd[s - 1] = td;
                    int   ti = bi[s]; bi[s] = bi[s - 1]; bi[s - 1] = ti;
                }
            }
        }
    }
    int* op = idxbuf + (size_t)((b * T_ + t) * NPTS + row) * KNN_;
#pragma unroll
    for (int s = 0; s < KNN_; s++) op[s] = bi[s];
}

__global__ void zero_kernel(float* p, int nelem) {
    const int i = blockIdx.x * blockDim.x + threadIdx.x;
    if (i < nelem) p[i] = 0.0f;
}

// out[b,t,0:4,n] = x[b,t,0:4,n]
__global__ void poscopy_kernel(const float* __restrict__ x, float* __restrict__ out) {
    const int i = blockIdx.x * blockDim.x + threadIdx.x;   // over B*T*4*N
    const int n = i & (NPTS - 1);
    const int rest = i >> 9;
    const int ch = rest & 3;
    const int bt = rest >> 2;
    out[((size_t)(bt * OUTC + ch)) * NPTS + n] = x[((size_t)(bt * CIN + ch)) * NPTS + n];
}

// ---------------------------------------------------------------------------
// One LSTM timestep, fully fused:
//   gather comb tile -> LDS, W tile -> LDS, f32 WMMA GEMM (14 x 7 k-steps),
//   bias + gate staging in LDS, LSTM elementwise + max over K neighbors.
// Grid: (128 col-tiles, 8 hid-tiles, B). Block: 256 threads = 8 waves.
// Wave w owns W-rows [16w,16w+16) of the 128-row {i,f,o,g}x32 tile and all
// 4 column sub-tiles -> 4 v8f accumulators.
// ---------------------------------------------------------------------------
__global__ __launch_bounds__(256) void step_kernel(
    const float* __restrict__ x, const float* __restrict__ W,
    const float* __restrict__ bias, const int* __restrict__ idxbuf,
    const float* __restrict__ hprev, const float* __restrict__ cprev,
    float* __restrict__ hnext, float* __restrict__ cnext,
    float* __restrict__ out, int t)
{
    __shared__ float smem[SMEMF];
    const int tid  = threadIdx.x;
    const int wv   = tid >> 5;
    const int lane = tid & 31;
    const int colTile = blockIdx.x, hidTile = blockIdx.y, b = blockIdx.z;
    const int col0 = colTile * COLT;
    const int n0   = colTile * 4;
    const int hid0 = hidTile * HTILE;
    const int tprev = (t == 0) ? 0 : (t - 1);

    const float* xq = x + (size_t)((b * T_ + t)     * CIN) * NPTS;
    const float* xp = x + (size_t)((b * T_ + tprev) * CIN) * NPTS;
    const int* idxbt = idxbuf + (size_t)((b * T_ + t) * NPTS) * KNN_;
    const float* hprevb = hprev + (size_t)(b * HID) * NPTS;

    v8f acc0 = {}, acc1 = {}, acc2 = {}, acc3 = {};

    // A-tile load mapping: 2 threads per row, 14 contiguous k each.
    const int arow  = tid >> 1;
    const int akb   = (tid & 1) * 14;
    const int agrow = (arow >> 5) * HID + hid0 + (arow & 31);   // gate*256 + hid
    const float* wp = W + (size_t)agrow * CTOT + akb;

    // B-tile (gathered comb) mapping: 4 threads per column, 7 k each.
    const int bcol = tid >> 2;
    const int bkb  = (tid & 3) * 7;
    const int gcol = col0 + bcol;
    const int bn   = gcol >> 4;
    const int bkq  = gcol & 15;
    const int gix  = idxbt[(size_t)bn * KNN_ + bkq];

    for (int kc = 0; kc < NKC; kc++) {
        const int c0 = kc * KC;
        // ---- A tile: W rows for {i,f,o,g} x 32 hidden units
#pragma unroll
        for (int e = 0; e < 14; e++)
            smem[arow * ASTRIDE + akb + e] = wp[c0 + e];
        // ---- B tile: build comb[c, col] with on-the-fly gather
#pragma unroll
        for (int e = 0; e < 7; e++) {
            const int c = c0 + bkb + e;
            float v;
            if (c < CIN)              v = xq[(size_t)c * NPTS + bn];
            else if (c < CIN + 4) {   const int j = c - CIN;
                                      v = xp[(size_t)j * NPTS + gix] - xq[(size_t)j * NPTS + bn]; }
            else                      v = hprevb[(size_t)(c - (CIN + 4)) * NPTS + gix];
            smem[BSOFF + bcol * ASTRIDE + bkb + e] = v;
        }
        __syncthreads();

        // ---- 7 WMMA k-steps x 4 column sub-tiles
        const int rr  = (wv << 4) + (lane & 15);          // A row (M = lane)
        const int lk2 = (lane >> 4) << 1;                 // lanes 16-31 hold K+2/K+3
#pragma unroll
        for (int ks = 0; ks < 7; ks++) {
            const int koff = ks * 4 + lk2;
            const v2f a  = *(const v2f*)&smem[rr * ASTRIDE + koff];
            const int cb = BSOFF + (lane & 15) * ASTRIDE + koff;
            const v2f b0 = *(const v2f*)&smem[cb];
            const v2f b1 = *(const v2f*)&smem[cb + 16 * ASTRIDE];
            const v2f b2 = *(const v2f*)&smem[cb + 32 * ASTRIDE];
            const v2f b3 = *(const v2f*)&smem[cb + 48 * ASTRIDE];
            acc0 = __builtin_amdgcn_wmma_f32_16x16x4_f32(false, a, false, b0, (short)0, acc0, false, false);
            acc1 = __builtin_amdgcn_wmma_f32_16x16x4_f32(false, a, false, b1, (short)0, acc1, false, false);
            acc2 = __builtin_amdgcn_wmma_f32_16x16x4_f32(false, a, false, b2, (short)0, acc2, false, false);
            acc3 = __builtin_amdgcn_wmma_f32_16x16x4_f32(false, a, false, b3, (short)0, acc3, false, false);
        }
        __syncthreads();
    }

    // ---- stage gates (+bias) to LDS; C/D layout: VGPR vi -> M = vi + 8*(lane>=16)
#pragma unroll
    for (int vi = 0; vi < 8; vi++) {
        const int r  = (wv << 4) + vi + ((lane >> 4) << 3);
        const float bb = bias[(r >> 5) * HID + hid0 + (r & 31)];
        const int cl = lane & 15;
        smem[r * GSTRIDE + cl +  0] = acc0[vi] + bb;
        smem[r * GSTRIDE + cl + 16] = acc1[vi] + bb;
        smem[r * GSTRIDE + cl + 32] = acc2[vi] + bb;
        smem[r * GSTRIDE + cl + 48] = acc3[vi] + bb;
    }
    __syncthreads();

    // ---- LSTM elementwise + max over K (all 16 neighbors local to this WG)
    if (tid < 128) {
        const int hid_l = tid >> 2, nl = tid & 3;
        const int n = n0 + nl;
        const int* ip = idxbt + (size_t)n * KNN_;
        const float* cpv = cprev + (size_t)(b * HID + hid0 + hid_l) * NPTS;
        float hmax = -3.4e38f, cmax = -3.4e38f;
#pragma unroll
        for (int k = 0; k < KNN_; k++) {
            const int cl = (nl << 4) + k;
            const float gi = smem[(     hid_l) * GSTRIDE + cl];
            const float gf = smem[(32 + hid_l) * GSTRIDE + cl];
            const float go = smem[(64 + hid_l) * GSTRIDE + cl];
            const float gg = smem[(96 + hid_l) * GSTRIDE + cl];
            const float cg = cpv[ip[k]];
            const float cn = sigf(gf) * cg + sigf(gi) * tanhf_(gg);
            const float hn = sigf(go) * tanhf_(cn);
            hmax = fmaxf(hmax, hn);
            cmax = fmaxf(cmax, cn);
        }
        const int hidg = hid0 + hid_l;
        hnext[(size_t)(b * HID + hidg) * NPTS + n] = hmax;
        cnext[(size_t)(b * HID + hidg) * NPTS + n] = cmax;
        out[((size_t)((b * T_ + t) * OUTC + 4 + hidg)) * NPTS + n] = hmax;
    }
}

// ---------------------------------------------------------------------------
extern "C" void kernel_launch(void* const* d_in, const int* in_sizes, int n_in,
                              void* d_out, int out_size, void* d_ws, size_t ws_size,
                              hipStream_t stream) {
    const float* x    = (const float*)d_in[0];   // (B,T,132,N)
    const float* W    = (const float*)d_in[1];   // (1024,392)
    const float* bias = (const float*)d_in[2];   // (1024,)
    float* out = (float*)d_out;                  // (B,T,260,N)
    (void)in_sizes; (void)n_in; (void)out_size; (void)ws_size;

    char* ws = (char*)d_ws;
    int*   idxbuf = (int*)ws;                           // B*T*N*K ints  = 1 MB
    float* h0 = (float*)(ws + (size_t)1 * (1 << 20));   // B*HID*N       = 1 MB
    float* c0 = (float*)(ws + (size_t)2 * (1 << 20));
    float* h1 = (float*)(ws + (size_t)3 * (1 << 20));
    float* c1 = (float*)(ws + (size_t)4 * (1 << 20));

    knn_kernel<<<B_ * T_ * 2, 256, 0, stream>>>(x, idxbuf);

    // zero initial h0,c0 (contiguous 2 MB region)
    const int zn = 2 * B_ * HID * NPTS;
    zero_kernel<<<(zn + 255) / 256, 256, 0, stream>>>(h0, zn);

    poscopy_kernel<<<(B_ * T_ * 4 * NPTS) / 256, 256, 0, stream>>>(x, out);

    float* hb[2] = { h0, h1 };
    float* cb[2] = { c0, c1 };
    const dim3 grid(NPTS * KNN_ / COLT, HID / HTILE, B_);
    for (int t = 0; t < T_; t++) {
        step_kernel<<<grid, 256, 0, stream>>>(x, W, bias, idxbuf,
                                              hb[t & 1], cb[t & 1],
                                              hb[(t + 1) & 1], cb[(t + 1) & 1],
                                              out, t);
    }
}